// ChainCRF_7507602833854
// MI455X (gfx1250) — compile-verified
//
#include <hip/hip_runtime.h>
#include <math.h>

// ChainCRF forward NLL on MI455X (gfx1250, wave32).
//
// Recursion in linear space so the KxK contraction runs on the matrix units:
//   part_new[j] = m + log( (exp(part-m))^T @ exp(trans) )[j] + emb[id_t][j]
// exp(trans) (48x48, constant) lives in 72 VGPRs as 36 f32 WMMA B-fragments
// for the whole 512-step scan. One wave per 16-batch tile (4 workgroups).
//
// Latency engineering (512-long serial scan chain; bandwidth is noise):
//  * part state register-resident in the exact WMMA C/D layout; row max via
//    __shfl_xor tree instead of a serial LDS scan.
//  * emb rows for t+1 fetched with GLOBAL_LOAD_ASYNC_TO_LDS_B128 (ASYNCcnt,
//    zero VGPR footprint -- fixes the round-2 scratch spills) into a
//    double-buffered LDS tile; __builtin_prefetch warms L2 at distance 2.
//  * ids staged to LDS once; mask register-prefetched; target-path energy
//    keeps prev label in a register (independent of the scan chain).

typedef __attribute__((ext_vector_type(2))) float v2f;
typedef __attribute__((ext_vector_type(8))) float v8f;

#define BATCH   64
#define LSEQ    512
#define KST     48      // number of CRF states (NUM_LABELS + 1)
#define BT      16      // batch rows per workgroup (one WMMA M-tile)
#define NKT     12      // K-dim tiles of 4 (48/4)
#define NNT     3       // N-dim tiles of 16 (48/16)

// 96 B/lane global -> LDS async copy. The instruction's IOFFSET advances BOTH
// the LDS and the global address (CDNA5 ISA), so one base pair
// covers the whole chunk. Tracked by ASYNCcnt.
__device__ __forceinline__ void async_copy_96B(unsigned lds_off, const float* g) {
    asm volatile(
        "global_load_async_to_lds_b128 %0, %1, off\n\t"
        "global_load_async_to_lds_b128 %0, %1, off offset:16\n\t"
        "global_load_async_to_lds_b128 %0, %1, off offset:32\n\t"
        "global_load_async_to_lds_b128 %0, %1, off offset:48\n\t"
        "global_load_async_to_lds_b128 %0, %1, off offset:64\n\t"
        "global_load_async_to_lds_b128 %0, %1, off offset:80"
        :: "v"(lds_off), "v"(g) : "memory");
}
__device__ __forceinline__ void wait_async0() {
    asm volatile("s_wait_asynccnt 0x0" ::: "memory");
}
// Generic LDS pointer -> 32-bit LDS byte address (flat LDS aperture keeps the
// DS offset in the low 32 bits, ISA 10.2 aperture table).
__device__ __forceinline__ unsigned lds_addr(const void* p) {
    return (unsigned)(unsigned long long)p;
}

__global__ __launch_bounds__(32) void crf_nll_kernel(
    const int*   __restrict__ ids,      // [B, L]
    const int*   __restrict__ target,   // [B, L]
    const float* __restrict__ mask,     // [B, L]
    const float* __restrict__ emb,      // [V, K]
    const float* __restrict__ trans,    // [K, K]
    float*       __restrict__ out)      // [B]
{
    __shared__ __align__(16) float s_trans[KST][KST];   // raw trans
    __shared__ __align__(16) float s_p[BT][KST];        // exp(part - m)
    __shared__ __align__(16) float s_e[2][BT][KST];     // emb rows, dbl-buffered
    __shared__ __align__(16) int   s_idsv[BT][LSEQ];    // staged input_ids
    __shared__ float s_mk[2][BT];                       // mask, dbl-buffered
    __shared__ float s_lz[BT];                          // final logZ per row

    const int lane = threadIdx.x;       // 0..31 (wave32)
    const int half = lane >> 4;         // 0: lanes 0-15, 1: lanes 16-31
    const int l15  = lane & 15;
    const int b0   = blockIdx.x * BT;

    // ---- stage all ids for this batch tile (contiguous int4 copy) ----
    {
        const int4* gi = (const int4*)(ids + b0 * LSEQ);
        int4*       di = (int4*)&s_idsv[0][0];
        for (int i = lane; i < BT * LSEQ / 4; i += 32) di[i] = gi[i];
    }
    // ---- raw trans into LDS (target-path energy, row K-1 for init) ----
    for (int idx = lane; idx < KST * KST; idx += 32)
        s_trans[idx / KST][idx % KST] = trans[idx];

    // ---- 36 B-fragments of exp(trans), straight from global into VGPRs ----
    // B 4x16 f32 layout: VGPR0 = K(2*half), VGPR1 = K(2*half+1), N = lane&15.
    v2f bfrag[NKT][NNT];
#pragma unroll
    for (int kt = 0; kt < NKT; ++kt)
#pragma unroll
        for (int nt = 0; nt < NNT; ++nt) {
            const int kr = kt * 4 + half * 2;
            const int nc = nt * 16 + l15;
            bfrag[kt][nt].x = __expf(trans[kr * KST + nc]);
            bfrag[kt][nt].y = __expf(trans[(kr + 1) * KST + nc]);
        }
    __syncthreads();    // single-wave group: S_NOP (DS is in-order per wave)

    // Each lane's coalesced emb chunk: row = lane/2, 24 cols at (lane&1)*24.
    const int row_e = lane >> 1;
    const int colb  = (lane & 1) * 24;

    // ---- async-load e(t=0) into buffer 0, then wait ----
    {
        const int id = s_idsv[row_e][0];
        async_copy_96B(lds_addr(&s_e[0][row_e][colb]),
                       emb + (size_t)id * KST + colb);
    }
    wait_async0();

    float mkreg = 0.0f;
    if (lane < BT) {
        mkreg = mask[(b0 + lane) * LSEQ + 0];
        s_mk[0][lane] = mkreg;
        mkreg = mask[(b0 + lane) * LSEQ + 1];   // prefetch t=1
    }

    // part state: lane owns rows r+8*half (r=0..7), cols {l15, 16+l15, 32+l15}
    // == exactly the f32 WMMA 16x16 C/D layout.
    float part_reg[8][3];
    float m_reg[8];
    float tgt_acc = 0.0f;
    int   prev_tg = KST - 1;

    // ---- t = 0: part0 = (trans[K-1, j] + e[b, j]) * mask ----
#pragma unroll
    for (int r = 0; r < 8; ++r) {
        const int row = r + 8 * half;
        const float mk = s_mk[0][row];
#pragma unroll
        for (int j = 0; j < 3; ++j) {
            const int col = j * 16 + l15;
            part_reg[r][j] = (s_trans[KST - 1][col] + s_e[0][row][col]) * mk;
        }
    }
    if (lane < BT) {
        const int tg = target[(b0 + lane) * LSEQ + 0];
        tgt_acc = (s_trans[KST - 1][tg] + s_e[0][lane][tg]) * s_mk[0][lane];
        prev_tg = tg;
    }

    // ---- issue async e(t=1) into buffer 1 (overlaps the tail below) ----
    {
        const int id = s_idsv[row_e][1];
        async_copy_96B(lds_addr(&s_e[1][row_e][colb]),
                       emb + (size_t)id * KST + colb);
    }

    // row max via shfl-xor tree + store p = exp(part - m) to LDS
    auto reduce_and_store_p = [&]() {
#pragma unroll
        for (int r = 0; r < 8; ++r) {
            float m = fmaxf(part_reg[r][0], fmaxf(part_reg[r][1], part_reg[r][2]));
            m = fmaxf(m, __shfl_xor(m, 1, 32));
            m = fmaxf(m, __shfl_xor(m, 2, 32));
            m = fmaxf(m, __shfl_xor(m, 4, 32));
            m = fmaxf(m, __shfl_xor(m, 8, 32));
            m_reg[r] = m;
            const int row = r + 8 * half;
#pragma unroll
            for (int j = 0; j < 3; ++j)
                s_p[row][j * 16 + l15] = __expf(part_reg[r][j] - m);
        }
    };
    reduce_and_store_p();

    // ================= 511 sequential scan steps =================
    // Invariant at loop entry: async loads for e(t) are in flight -> s_e[t&1].
    for (int t = 1; t < LSEQ; ++t) {
        const int buf = t & 1;

        // ---- (16x48) @ (48x48) on the matrix units (independent of s_e) ----
        v8f acc0 = {}, acc1 = {}, acc2 = {};
#pragma unroll
        for (int kt = 0; kt < NKT; ++kt) {
            const int kc = kt * 4 + half * 2;     // A: row=l15, K=kc..kc+1
            const v2f a = *(const v2f*)&s_p[l15][kc];
            acc0 = __builtin_amdgcn_wmma_f32_16x16x4_f32(
                       false, a, false, bfrag[kt][0], (short)0, acc0, false, false);
            acc1 = __builtin_amdgcn_wmma_f32_16x16x4_f32(
                       false, a, false, bfrag[kt][1], (short)0, acc1, false, false);
            acc2 = __builtin_amdgcn_wmma_f32_16x16x4_f32(
                       false, a, false, bfrag[kt][2], (short)0, acc2, false, false);
        }

        // e(t) now guaranteed resident in s_e[buf]
        wait_async0();

        // commit mask(t), register-prefetch mask(t+1)
        if (lane < BT) {
            s_mk[buf][lane] = mkreg;
            if (t + 1 < LSEQ) mkreg = mask[(b0 + lane) * LSEQ + t + 1];
        }

        // issue async e(t+1) into the other buffer; L2 warm-up at t+2
        if (t + 1 < LSEQ) {
            const int id = s_idsv[row_e][t + 1];
            async_copy_96B(lds_addr(&s_e[buf ^ 1][row_e][colb]),
                           emb + (size_t)id * KST + colb);
        }
        if (t + 2 < LSEQ)   // -> global_prefetch_b8
            __builtin_prefetch(emb + (size_t)s_idsv[row_e][t + 2] * KST + colb, 0, 1);

        // ---- back to log space, blend by mask (register-resident part) ----
#pragma unroll
        for (int r = 0; r < 8; ++r) {
            const int row  = r + 8 * half;
            const float mk = s_mk[buf][row];
            const float mp = m_reg[r];
            {
                const float val = __logf(acc0[r]) + mp + s_e[buf][row][l15];
                part_reg[r][0] += (val - part_reg[r][0]) * mk;
            }
            {
                const float val = __logf(acc1[r]) + mp + s_e[buf][row][16 + l15];
                part_reg[r][1] += (val - part_reg[r][1]) * mk;
            }
            {
                const float val = __logf(acc2[r]) + mp + s_e[buf][row][32 + l15];
                part_reg[r][2] += (val - part_reg[r][2]) * mk;
            }
        }

        reduce_and_store_p();

        // ---- target-path energy (independent of the scan chain) ----
        if (lane < BT) {
            const int tg = target[(b0 + lane) * LSEQ + t];
            tgt_acc += (s_trans[prev_tg][tg] + s_e[buf][lane][tg]) * s_mk[buf][lane];
            prev_tg = tg;
        }
    }

    // ---- final logsumexp over states via shfl-xor sum tree ----
#pragma unroll
    for (int r = 0; r < 8; ++r) {
        float s = __expf(part_reg[r][0] - m_reg[r])
                + __expf(part_reg[r][1] - m_reg[r])
                + __expf(part_reg[r][2] - m_reg[r]);
        s += __shfl_xor(s, 1, 32);
        s += __shfl_xor(s, 2, 32);
        s += __shfl_xor(s, 4, 32);
        s += __shfl_xor(s, 8, 32);
        if (l15 == 0) s_lz[r + 8 * half] = m_reg[r] + __logf(s);
    }
    __syncthreads();
    if (lane < BT)
        out[b0 + lane] = s_lz[lane] - tgt_acc;
}

extern "C" void kernel_launch(void* const* d_in, const int* in_sizes, int n_in,
                              void* d_out, int out_size, void* d_ws, size_t ws_size,
                              hipStream_t stream) {
    (void)in_sizes; (void)n_in; (void)d_ws; (void)ws_size; (void)out_size;
    const int*   ids    = (const int*)  d_in[0];
    const int*   target = (const int*)  d_in[1];
    const float* mask   = (const float*)d_in[2];
    const float* emb    = (const float*)d_in[3];
    const float* trans  = (const float*)d_in[4];
    float*       out    = (float*)d_out;

    crf_nll_kernel<<<BATCH / BT, 32, 0, stream>>>(ids, target, mask, emb, trans, out);
}